// GraphConvolution_7224134992249
// MI455X (gfx1250) — compile-verified
//
#include <hip/hip_runtime.h>
#include <hip/hip_bf16.h>

#define N_NODES  100000
#define N_EDGES  1600000
#define IN_FEAT  128
#define OUT_FEAT 64

typedef __attribute__((ext_vector_type(2))) float v2f;
typedef __attribute__((ext_vector_type(8))) float v8f;

// ---------------------------------------------------------------------------
// Kernel 1: support = X @ W using V_WMMA_F32_16X16X4_F32 (exact fp32 tensor op)
// One wave32 computes a 16x64 tile of the output (4 WMMA accumulators).
// N_NODES = 6250 * 16 exactly, so no remainder handling needed.
//
// VGPR layouts per CDNA5 ISA 7.12.2:
//   A (16x4, MxK): lanes 0-15 -> M=lane, {K=k,K=k+1}; lanes 16-31 -> M=lane-16,
//                  {K=k+2,K=k+3}
//   B (4x16, KxN): lanes 0-15 -> N=lane, {K=k,K=k+1}; lanes 16-31 -> N=lane-16,
//                  {K=k+2,K=k+3}
//   C/D (16x16):   VGPR i -> M=i (lanes 0-15) / M=i+8 (lanes 16-31), N=lane%16
// ---------------------------------------------------------------------------
__global__ __launch_bounds__(256) void gcn_gemm_wmma(const float* __restrict__ X,
                                                     const float* __restrict__ W,
                                                     float* __restrict__ S) {
  const int waveId = (int)((blockIdx.x * blockDim.x + threadIdx.x) >> 5);
  const int lane   = (int)(threadIdx.x & 31);
  const int nTiles = N_NODES / 16;  // 6250
  if (waveId >= nTiles) return;     // wave-uniform exit (EXEC all-0 waves skip)

  const int row0 = waveId * 16;
  const int m    = lane & 15;          // M (for A) / N-in-tile (for B, C/D)
  const int kh   = (lane >> 4) << 1;   // 0 for lanes 0-15, 2 for lanes 16-31

  v8f c0 = {}, c1 = {}, c2 = {}, c3 = {};

  const float* xrow = X + (size_t)(row0 + m) * IN_FEAT + kh;

#pragma unroll 4
  for (int k = 0; k < IN_FEAT; k += 4) {
    v2f a;
    a.x = xrow[k];
    a.y = xrow[k + 1];

    const float* wb = W + (size_t)(k + kh) * OUT_FEAT + m;
    v2f b0, b1, b2, b3;
    b0.x = wb[0];  b0.y = wb[OUT_FEAT + 0];
    b1.x = wb[16]; b1.y = wb[OUT_FEAT + 16];
    b2.x = wb[32]; b2.y = wb[OUT_FEAT + 32];
    b3.x = wb[48]; b3.y = wb[OUT_FEAT + 48];

    c0 = __builtin_amdgcn_wmma_f32_16x16x4_f32(false, a, false, b0, (short)0, c0, false, false);
    c1 = __builtin_amdgcn_wmma_f32_16x16x4_f32(false, a, false, b1, (short)0, c1, false, false);
    c2 = __builtin_amdgcn_wmma_f32_16x16x4_f32(false, a, false, b2, (short)0, c2, false, false);
    c3 = __builtin_amdgcn_wmma_f32_16x16x4_f32(false, a, false, b3, (short)0, c3, false, false);
  }

  // Write back the 16x64 tile.
  const int mBase = row0 + ((lane >> 4) << 3);  // +8 rows for lanes 16-31
#pragma unroll
  for (int i = 0; i < 8; ++i) {
    float* srow = S + (size_t)(mBase + i) * OUT_FEAT + m;
    srow[0]  = c0[i];
    srow[16] = c1[i];
    srow[32] = c2[i];
    srow[48] = c3[i];
  }
}

// ---------------------------------------------------------------------------
// Kernel 2: out[r][c] = bias[c]   (also initializes out before the atomics)
// ---------------------------------------------------------------------------
__global__ __launch_bounds__(256) void gcn_init_bias(const float* __restrict__ bias,
                                                     float* __restrict__ out) {
  const int i = (int)(blockIdx.x * blockDim.x + threadIdx.x);
  if (i < N_NODES * OUT_FEAT) {
    out[i] = bias[i & (OUT_FEAT - 1)];
  }
}

// ---------------------------------------------------------------------------
// Kernel 3: out[row[e]] += val[e] * support[col[e]]   (L2-side f32 atomics)
// 16 threads per edge; each thread handles a 16-byte (float4) chunk of the
// 64-float feature row. The 16 replicated index loads hit the same address
// and coalesce into one request; actual DRAM index traffic stays 12 B/edge.
// ---------------------------------------------------------------------------
__global__ __launch_bounds__(256) void gcn_spmm_scatter(const int*   __restrict__ arow,
                                                        const int*   __restrict__ acol,
                                                        const float* __restrict__ avals,
                                                        const float* __restrict__ S,
                                                        float*       __restrict__ out) {
  const long long t = (long long)blockIdx.x * blockDim.x + threadIdx.x;
  const int e = (int)(t >> 4);
  if (e >= N_EDGES) return;
  const int c = ((int)t & 15) << 2;  // chunk offset in floats: 0,4,...,60

  const int   r  = arow[e];
  const int   cl = acol[e];
  const float v  = avals[e];

  const float4 g = *(const float4*)(S + (size_t)cl * OUT_FEAT + c);
  float* dst = out + (size_t)r * OUT_FEAT + c;

  // unsafeAtomicAdd lowers to global_atomic_add_f32 (no CAS loop, no return).
  unsafeAtomicAdd(dst + 0, v * g.x);
  unsafeAtomicAdd(dst + 1, v * g.y);
  unsafeAtomicAdd(dst + 2, v * g.z);
  unsafeAtomicAdd(dst + 3, v * g.w);
}

// ---------------------------------------------------------------------------
extern "C" void kernel_launch(void* const* d_in, const int* in_sizes, int n_in,
                              void* d_out, int out_size, void* d_ws, size_t ws_size,
                              hipStream_t stream) {
  const float* X     = (const float*)d_in[0];  // [N_NODES, IN_FEAT]
  const int*   arow  = (const int*)  d_in[1];  // [N_EDGES]
  const int*   acol  = (const int*)  d_in[2];  // [N_EDGES]
  const float* avals = (const float*)d_in[3];  // [N_EDGES]
  const float* W     = (const float*)d_in[4];  // [IN_FEAT, OUT_FEAT]
  const float* bias  = (const float*)d_in[5];  // [OUT_FEAT]
  float*       out   = (float*)d_out;          // [N_NODES, OUT_FEAT]
  float*       supp  = (float*)d_ws;           // [N_NODES, OUT_FEAT] scratch (25.6 MB)

  // 1) support = X @ W : 6250 wave-tiles, 8 waves (256 threads) per block.
  {
    const int nTiles = N_NODES / 16;                  // 6250
    const int wavesPerBlock = 256 / 32;               // 8
    const int grid = (nTiles + wavesPerBlock - 1) / wavesPerBlock;  // 782
    gcn_gemm_wmma<<<grid, 256, 0, stream>>>(X, W, supp);
  }

  // 2) out = broadcast(bias)
  {
    const int total = N_NODES * OUT_FEAT;             // 6.4M
    const int grid = (total + 255) / 256;             // 25000
    gcn_init_bias<<<grid, 256, 0, stream>>>(bias, out);
  }

  // 3) out[row] += val * support[col]
  {
    const long long total = (long long)N_EDGES * 16;  // 25.6M threads
    const int grid = (int)((total + 255) / 256);      // 100000
    gcn_spmm_scatter<<<grid, 256, 0, stream>>>(arow, acol, avals, supp, out);
  }
}